// SimpleContrastiveLoss_23295902613646
// MI455X (gfx1250) — compile-verified
//
#include <hip/hip_runtime.h>
#include <math.h>

// Problem constants (from reference setup_inputs)
#define B_N 4096
#define K_N 128
#define D_N 512
#define INV_T (1.0f / 0.07f)
#define EPS_SUM 1e-10f
#define EPS_NORM 1e-12f

typedef __attribute__((ext_vector_type(2))) float v2f;
typedef __attribute__((ext_vector_type(8))) float v8f;

__device__ __forceinline__ float waveReduceSum(float v) {
#pragma unroll
    for (int off = 16; off > 0; off >>= 1)
        v += __shfl_xor(v, off, 32);
    return v;
}

__global__ void init_ws_kernel(float* ws) {
    if (threadIdx.x < 2) ws[threadIdx.x] = 0.0f;
}

__global__ void finalize_kernel(const float* __restrict__ ws, float* __restrict__ out) {
    if (threadIdx.x == 0) {
        float total = ws[0];
        float n     = ws[1];
        out[0] = (n > 0.0f) ? (total / fmaxf(n, 1.0f)) : 0.0f;
    }
}

// One workgroup per sample b. 8 waves; wave w owns k-rows [16w, 16w+16).
// Dot products computed with V_WMMA_F32_16X16X4_F32:
//   A (16x4 fp32, documented layout) = 16 retrieved rows x 4 d-values
//   B (4x16 fp32) = anchor d-chunk replicated across all 16 columns
//   => every column of C = the 16 desired dot products (fp32 accumulate).
__global__ __launch_bounds__(256) void contrastive_main_kernel(
    const float* __restrict__ feat,      // [B, D]
    const int*   __restrict__ labels,    // [B]
    const float* __restrict__ rfeat,     // [B, K, D]
    const int*   __restrict__ rlabels,   // [B, K]
    float* __restrict__ ws)              // [0]=total loss, [1]=valid count
{
    __shared__ float s_a[D_N];       // anchor vector
    __shared__ float s_dot[K_N];     // raw dot products
    __shared__ float s_rs[K_N];      // ||r_k||^2
    __shared__ float s_red[3][8];    // per-wave reduction partials
    __shared__ float s_fss;          // ||f||^2

    const int b    = blockIdx.x;
    const int tid  = threadIdx.x;
    const int lane = tid & 31;
    const int wave = tid >> 5;

    // ---- Stage anchor into LDS, compute ||f||^2 ----
    const float* fp = feat + (size_t)b * D_N;
    v2f fv = *(const v2f*)(fp + 2 * tid);
    s_a[2 * tid]     = fv.x;
    s_a[2 * tid + 1] = fv.y;
    float fss = fv.x * fv.x + fv.y * fv.y;
    fss = waveReduceSum(fss);
    if (lane == 0) s_red[0][wave] = fss;
    __syncthreads();
    if (tid == 0) {
        float t = 0.0f;
#pragma unroll
        for (int i = 0; i < 8; ++i) t += s_red[0][i];
        s_fss = t;
    }
    __syncthreads();

    // ---- WMMA GEMV: dot[k] = sum_d r[b,k,d] * f[b,d], plus ||r_k||^2 ----
    const int h  = lane >> 4;   // half-wave: selects K-pair {2h, 2h+1}
    const int m  = lane & 15;   // A-matrix row M
    const int k0 = wave * 16;

    const float* rp = rfeat + ((size_t)b * K_N + (size_t)(k0 + m)) * D_N + 2 * h;
    const float* ap = s_a + 2 * h;

    v8f   acc = {};
    float rss = 0.0f;

#pragma unroll 4
    for (int s = 0; s < D_N / 4; ++s) {
        // A operand: lane holds r[k0+m, d0 + 2h + {0,1}]  (d0 = 4s)
        v2f av = __builtin_nontemporal_load((const v2f*)(rp + 4 * s));
        // B operand: anchor values a[d0 + 2h + {0,1}], identical for all 16 columns
        v2f bv = *(const v2f*)(ap + 4 * s);
        rss += av.x * av.x + av.y * av.y;
        acc = __builtin_amdgcn_wmma_f32_16x16x4_f32(
            /*neg_a=*/false, av, /*neg_b=*/false, bv,
            /*c_mod=*/(short)0, acc, /*reuse_a=*/false, /*reuse_b=*/false);
    }

    // lane L covers d == {2h,2h+1} mod 4; combine with partner lane (L xor 16)
    float rss_full = rss + __shfl_xor(rss, 16, 32);
    if (h == 0) s_rs[k0 + m] = rss_full;   // lanes 0..15 -> rows k0..k0+15

    // C layout: vgpr c, lanes 0-15 -> M=c ; lanes 16-31 -> M=c+8 (cols identical)
    if (m == 0) {
#pragma unroll
        for (int c = 0; c < 8; ++c)
            s_dot[k0 + c + 8 * h] = acc[c];
    }
    __syncthreads();

    // ---- Per-k softmax terms (threads 0..127 each handle one k) ----
    float e_all = 0.0f, e_pos = 0.0f, n_pos = 0.0f;
    if (tid < K_N) {
        const int k   = tid;
        const float d = s_dot[k];
        const float nf = fmaxf(sqrtf(s_fss), EPS_NORM);
        const float nr = fmaxf(sqrtf(s_rs[k]), EPS_NORM);
        const float sim = (d / (nf * nr)) * INV_T;
        const int rl  = rlabels[(size_t)b * K_N + k];
        const int lbl = labels[b];
        const float validf = (rl != -1) ? 1.0f : 0.0f;
        const float posf   = (rl == lbl) ? 1.0f : 0.0f;
        e_all = expf(sim) * validf;
        e_pos = e_all * posf;
        n_pos = posf * validf;
    }

    float r0 = waveReduceSum(e_all);
    float r1 = waveReduceSum(e_pos);
    float r2 = waveReduceSum(n_pos);
    if (lane == 0) {
        s_red[0][wave] = r0;
        s_red[1][wave] = r1;
        s_red[2][wave] = r2;
    }
    __syncthreads();

    if (tid == 0) {
        float all_sum = EPS_SUM, pos_sum = EPS_SUM, cnt = 0.0f;
#pragma unroll
        for (int i = 0; i < 8; ++i) {
            all_sum += s_red[0][i];
            pos_sum += s_red[1][i];
            cnt     += s_red[2][i];
        }
        const float loss = -logf(pos_sum / all_sum);
        const bool finite = !__builtin_isnan(loss) && !__builtin_isinf(loss);
        if ((cnt > 0.0f) && finite) {
            atomicAdd(&ws[0], loss);
            atomicAdd(&ws[1], 1.0f);
        }
    }
}

extern "C" void kernel_launch(void* const* d_in, const int* in_sizes, int n_in,
                              void* d_out, int out_size, void* d_ws, size_t ws_size,
                              hipStream_t stream) {
    const float* feat    = (const float*)d_in[0];
    const int*   labels  = (const int*)d_in[1];
    const float* rfeat   = (const float*)d_in[2];
    const int*   rlabels = (const int*)d_in[3];
    float* ws  = (float*)d_ws;
    float* out = (float*)d_out;

    init_ws_kernel<<<1, 32, 0, stream>>>(ws);
    contrastive_main_kernel<<<B_N, 256, 0, stream>>>(feat, labels, rfeat, rlabels, ws);
    finalize_kernel<<<1, 32, 0, stream>>>(ws, out);
}